// PredictionDecoder_64381559767225
// MI455X (gfx1250) — compile-verified
//
#include <hip/hip_runtime.h>

#define N_USERC 60082
#define DC 128
#define BC 8
#define KC 32

typedef __attribute__((ext_vector_type(2))) float v2f;
typedef __attribute__((ext_vector_type(8))) float v8f;

// workspace layout (floats): [0,128) v ; [128,136) base[b] ; [136] c ; [137] c2 ; [138,202) u2

__global__ void prep_kernel(const float* __restrict__ user_embedding,
                            const float* __restrict__ user_emb_table,
                            const float* __restrict__ proj_w,
                            const float* __restrict__ proj_b,
                            const float* __restrict__ theta,
                            const float* __restrict__ w_his2,
                            const float* __restrict__ b_his2,
                            const float* __restrict__ w_st,
                            const float* __restrict__ b_st,
                            const float* __restrict__ w_u,
                            const float* __restrict__ b_u,
                            const int*   __restrict__ user_id,
                            float* __restrict__ ws)
{
    int t = threadIdx.x;                 // 128 threads
    float acc = 0.f;
    for (int d = 0; d < DC; ++d) acc += proj_w[t * DC + d] * w_st[d];
    ws[t] = acc;                          // v = proj_w @ w_st
    if (t < 64) {
        float a2 = 0.f;
        for (int d = 0; d < DC; ++d) a2 += w_his2[t * DC + d] * w_st[d];
        ws[138 + t] = a2;                 // u2 = w_his2 @ w_st
    }
    if (t < BC) {
        int uid = user_id[t];
        float th = theta[uid];
        float a3 = 0.f;
        for (int d = 0; d < DC; ++d)
            a3 += ((1.f - th) * user_embedding[t * DC + d] +
                   th * user_emb_table[(size_t)uid * DC + d]) * w_u[d];
        ws[128 + t] = a3 + b_u[0] + b_st[0];   // base[b]
    }
    if (t == 0) {
        float cc = 0.f, cc2 = 0.f;
        for (int d = 0; d < DC; ++d) { cc += proj_b[d] * w_st[d]; cc2 += b_his2[d] * w_st[d]; }
        ws[136] = cc; ws[137] = cc2;
    }
}

// Each wave computes s[t..t+15] = table[t..t+15,:] @ v via V_WMMA_F32_16X16X4_F32
// (B-matrix columns all replicate v, so every C/D column equals s), then writes
// out[b, t..t+15] = s + base[b] for all 8 b.
__global__ void __launch_bounds__(256) main_kernel(const float* __restrict__ table,
                                                   const float* __restrict__ ws,
                                                   float* __restrict__ out)
{
    __shared__ float vsh[DC];
    int tid = threadIdx.x;
    if (tid < DC) vsh[tid] = ws[tid];
    __syncthreads();

    int lane = tid & 31;
    int wave = tid >> 5;
    int tile = blockIdx.x * 8 + wave;
    int t = tile * 16;

    int half = lane >> 4;                 // K sub-block / M 8..15 half
    int mrow = lane & 15;
    int row = t + mrow; if (row >= N_USERC) row = N_USERC - 1;   // clamp tail loads
    const float* arow = table + (size_t)row * DC + 2 * half;
    const float* brow = vsh + 2 * half;

    v8f acc = {};
    #pragma unroll
    for (int k = 0; k < DC; k += 4) {
        v2f a  = *(const v2f*)(arow + k);   // A: V0=K+2h, V1=K+1+2h per lane
        v2f bv = *(const v2f*)(brow + k);   // B: identical K pattern, replicated over N
        acc = __builtin_amdgcn_wmma_f32_16x16x4_f32(false, a, false, bv,
                                                    (short)0, acc, false, false);
    }

    // lanes 0-15 hold s[t+0..7] in acc[0..7]; lanes 16-31 hold s[t+8..15]
    int L2 = lane & 15;
    int b  = L2 >> 1;          // batch row 0..7
    int h  = L2 & 1;           // which half of acc
    int n0 = t + half * 8 + h * 4;
    float basev = ws[128 + b];

    float o0 = h ? acc[4] : acc[0];
    float o1 = h ? acc[5] : acc[1];
    float o2 = h ? acc[6] : acc[2];
    float o3 = h ? acc[7] : acc[3];

    size_t off = (size_t)b * N_USERC + n0;   // 8-byte aligned (N even, n0 % 4 == 0)
    if (n0 + 4 <= N_USERC) {
        v2f s0 = { o0 + basev, o1 + basev };
        v2f s1 = { o2 + basev, o3 + basev };
        *(v2f*)(out + off)     = s0;
        *(v2f*)(out + off + 2) = s1;
    } else {
        float o[4] = { o0, o1, o2, o3 };
        for (int i = 0; i < 4; ++i)
            if (n0 + i < N_USERC) out[off + i] = o[i] + basev;
    }
}

// One block per batch row b. 8 waves compute the 64 node deltas cooperatively,
// thread 0 applies them serially (deterministic; duplicate nodes accumulate
// like jnp .at[].add, the -alpha*s term fires once per distinct node).
__global__ void __launch_bounds__(256) fixup_kernel(const float* __restrict__ station_embedding,
                                                    const float* __restrict__ raw_field_embed,
                                                    const float* __restrict__ table,
                                                    const float* __restrict__ alpha,
                                                    const float* __restrict__ w_his1,
                                                    const float* __restrict__ b_his1,
                                                    const float* __restrict__ w_st,
                                                    const int*   __restrict__ his_nodes,
                                                    const int*   __restrict__ now_nodes,
                                                    const float* __restrict__ ws,
                                                    float* __restrict__ out)
{
    __shared__ float sdelta[2 * KC];
    __shared__ int   snode [2 * KC];
    int b    = blockIdx.x;
    int lane = threadIdx.x & 31;
    int wave = threadIdx.x >> 5;     // 8 waves
    const int* nowb = now_nodes + b * KC;
    const int* hisb = his_nodes + b * KC;

    for (int j = wave; j < 2 * KC; j += 8) {
        bool isNow = (j < KC);
        int node = isNow ? nowb[j] : hisb[j - KC];

        bool first = true;                       // uniform per wave
        for (int j2 = 0; j2 < j; ++j2) {
            int n2 = (j2 < KC) ? nowb[j2] : hisb[j2 - KC];
            if (n2 == node) { first = false; break; }
        }

        float al = alpha[node];

        // s[node] = table[node,:] . v + c  (wave-cooperative)
        float p = 0.f;
        const float* tr = table + (size_t)node * DC;
        for (int i = 0; i < 4; ++i) { int d = lane * 4 + i; p += tr[d] * ws[d]; }
        for (int off = 16; off; off >>= 1) p += __shfl_xor(p, off, 32);
        float s_n = p + ws[136];

        float delta = first ? (-al * s_n) : 0.f;

        if (isNow) {
            float q = 0.f;
            const float* er = station_embedding + (size_t)node * DC;
            for (int i = 0; i < 4; ++i) { int d = lane * 4 + i; q += er[d] * w_st[d]; }
            for (int off = 16; off; off >>= 1) q += __shfl_xor(q, off, 32);
            delta += al * q;
        } else {
            // alpha * ( lrelu(raw @ w_his1 + b_his1) . u2 + c2 )
            const float* rr = raw_field_embed + (size_t)node * DC;
            float part = 0.f;
            for (int mo = 0; mo < 2; ++mo) {
                int m = lane + mo * 32;
                float g = b_his1[m];
                for (int d = 0; d < DC; ++d) g += rr[d] * w_his1[d * 64 + m];
                g = (g > 0.f) ? g : 0.01f * g;        // jax.nn.leaky_relu default slope
                part += g * ws[138 + m];
            }
            for (int off = 16; off; off >>= 1) part += __shfl_xor(part, off, 32);
            delta += al * (part + ws[137]);
        }

        if (lane == 0) { sdelta[j] = delta; snode[j] = node; }
    }
    __syncthreads();
    if (threadIdx.x == 0) {
        float* ob = out + (size_t)b * N_USERC;
        for (int j = 0; j < 2 * KC; ++j) ob[snode[j]] += sdelta[j];
    }
}

extern "C" void kernel_launch(void* const* d_in, const int* in_sizes, int n_in,
                              void* d_out, int out_size, void* d_ws, size_t ws_size,
                              hipStream_t stream) {
    const float* user_embedding    = (const float*)d_in[0];
    const float* station_embedding = (const float*)d_in[1];
    const float* raw_field_embed   = (const float*)d_in[2];
    const int*   his_nodes         = (const int*)  d_in[3];
    const int*   now_nodes         = (const int*)  d_in[4];
    const int*   user_id           = (const int*)  d_in[5];
    const float* user_emb_table    = (const float*)d_in[6];
    const float* station_emb_table = (const float*)d_in[7];
    const float* proj_w            = (const float*)d_in[8];
    const float* proj_b            = (const float*)d_in[9];
    const float* theta             = (const float*)d_in[10];
    const float* alpha             = (const float*)d_in[11];
    const float* w_his1            = (const float*)d_in[12];
    const float* b_his1            = (const float*)d_in[13];
    const float* w_his2            = (const float*)d_in[14];
    const float* b_his2            = (const float*)d_in[15];
    const float* w_st              = (const float*)d_in[16];
    const float* b_st              = (const float*)d_in[17];
    const float* w_u               = (const float*)d_in[18];
    const float* b_u               = (const float*)d_in[19];
    float* out = (float*)d_out;
    float* ws  = (float*)d_ws;

    prep_kernel<<<1, 128, 0, stream>>>(user_embedding, user_emb_table, proj_w, proj_b,
                                       theta, w_his2, b_his2, w_st, b_st, w_u, b_u,
                                       user_id, ws);

    int tiles  = (N_USERC + 15) / 16;      // 3756
    int blocks = (tiles + 7) / 8;          // 470 blocks x 8 waves
    main_kernel<<<blocks, 256, 0, stream>>>(station_emb_table, ws, out);

    fixup_kernel<<<BC, 256, 0, stream>>>(station_embedding, raw_field_embed,
                                         station_emb_table, alpha, w_his1, b_his1,
                                         w_st, his_nodes, now_nodes, ws, out);
}